// FBGCN_Layer_24618752540870
// MI455X (gfx1250) — compile-verified
//
#include <hip/hip_runtime.h>

typedef __attribute__((ext_vector_type(2))) float v2f;
typedef __attribute__((ext_vector_type(4))) float v4f;
typedef __attribute__((ext_vector_type(8))) float v8f;
typedef __attribute__((ext_vector_type(4))) int   v4i;

#define NNODES   4096
#define FDIM     256
#define TILE_M   128
#define TILE_N   64
#define TILE_K   32
#define LDA      36   // A LDS row stride: mult-of-4 (b128-aligned), 36m % 64 distinct for m<16
#define LDB      72   // B LDS row stride: 2*72 % 64 == 16 -> half-waves hit disjoint banks

// ---------------------------------------------------------------------------
// Async global->LDS copy (CDNA5 GLOBAL_LOAD_ASYNC_TO_LDS_B128, ASYNCcnt),
// with a synchronous fallback if the toolchain lacks the builtins.
// Builtin signature (per hipcc diagnostic): (as1 v4i*, as3 v4i*, Imm, Imm).
// ---------------------------------------------------------------------------
#if __has_builtin(__builtin_amdgcn_global_load_async_to_lds_b128) && \
    __has_builtin(__builtin_amdgcn_s_wait_asynccnt)
__device__ __forceinline__ void async_copy16(float* lds, const float* gsrc) {
    __builtin_amdgcn_global_load_async_to_lds_b128(
        (__attribute__((address_space(1))) v4i*)gsrc,
        (__attribute__((address_space(3))) v4i*)lds,
        /*offset=*/0, /*cpol=*/0);
}
__device__ __forceinline__ void wait_async() {
    __builtin_amdgcn_s_wait_asynccnt(0);
}
#else
__device__ __forceinline__ void async_copy16(float* lds, const float* gsrc) {
    *(v4f*)lds = *(const v4f*)gsrc;
}
__device__ __forceinline__ void wait_async() {}
#endif

// ---------------------------------------------------------------------------
// C[M,N] = (A[M,K] @ B[K,N]), optional relu. Row-major. M%128==0, N%64==0,
// K%32==0. 256 threads = 8 waves; wave grid 4(M) x 2(N); 32x32 per wave via
// four V_WMMA_F32_16X16X4_F32 accumulators (full f32 precision path).
// Double-buffered LDS: async fill of stage s+1 overlaps WMMAs of stage s.
// ---------------------------------------------------------------------------
__global__ __launch_bounds__(256)
void gemm_f32_wmma(const float* __restrict__ A, const float* __restrict__ B,
                   float* __restrict__ C, int M, int N, int K, int relu)
{
    __shared__ float As[2][TILE_M * LDA];          // As[buf][row*LDA + k]
    __shared__ float Bs[2][TILE_K * LDB];          // Bs[buf][k*LDB + n] (row-major)

    const int tid   = threadIdx.x;
    const int lane  = tid & 31;
    const int wave  = tid >> 5;        // 0..7
    const int waveM = wave >> 1;       // 0..3
    const int waveN = wave & 1;        // 0..1

    const int blockM = blockIdx.x * TILE_M;
    const int blockN = blockIdx.y * TILE_N;

    const int l15   = lane & 15;
    const int khalf = lane >> 4;       // 0: lanes 0-15, 1: lanes 16-31

    v8f acc00 = {}, acc01 = {}, acc10 = {}, acc11 = {};

    // staging maps (b128 chunks, fully coalesced)
    const int arow0 = tid >> 3;        // A: rows tid/8 + 32*i, 8 chunks/row
    const int acol  = (tid & 7) * 4;
    const int brow0 = tid >> 4;        // B: rows tid/16 + 16*i, 16 chunks/row
    const int bcol  = (tid & 15) * 4;

    auto stage = [&](int buf, int kb) {
        #pragma unroll
        for (int i = 0; i < 4; ++i) {
            int r = arow0 + 32 * i;
            async_copy16(&As[buf][r * LDA + acol],
                         A + (size_t)(blockM + r) * K + kb + acol);
        }
        #pragma unroll
        for (int i = 0; i < 2; ++i) {
            int r = brow0 + 16 * i;
            async_copy16(&Bs[buf][r * LDB + bcol],
                         B + (size_t)(kb + r) * N + blockN + bcol);
        }
    };

    const int nStages = K / TILE_K;
    stage(0, 0);

    for (int s = 0; s < nStages; ++s) {
        const int cur = s & 1;
        wait_async();                       // this wave's fills of `cur` done
        __syncthreads();                    // all waves' fills of `cur` done
        if (s + 1 < nStages) stage(cur ^ 1, (s + 1) * TILE_K);  // overlaps compute

        #pragma unroll
        for (int kk = 0; kk < TILE_K; kk += 4) {
            const int ko = kk + 2 * khalf;
            // A frag 16x4: lanes 0-15 hold K={kk,kk+1}, lanes 16-31 K={kk+2,kk+3}
            v2f a0 = *(const v2f*)&As[cur][(waveM * 32 +      l15) * LDA + ko];
            v2f a1 = *(const v2f*)&As[cur][(waveM * 32 + 16 + l15) * LDA + ko];
            // B frag 4x16: VGPR0 = row K=ko, VGPR1 = row K=ko+1, striped by N
            const float* bp0 = &Bs[cur][ko * LDB + waveN * 32 + l15];
            const float* bp1 = bp0 + 16;
            v2f b0 = { bp0[0], bp0[LDB] };
            v2f b1 = { bp1[0], bp1[LDB] };
            acc00 = __builtin_amdgcn_wmma_f32_16x16x4_f32(false, a0, false, b0, (short)0, acc00, false, false);
            acc01 = __builtin_amdgcn_wmma_f32_16x16x4_f32(false, a0, false, b1, (short)0, acc01, false, false);
            acc10 = __builtin_amdgcn_wmma_f32_16x16x4_f32(false, a1, false, b0, (short)0, acc10, false, false);
            acc11 = __builtin_amdgcn_wmma_f32_16x16x4_f32(false, a1, false, b1, (short)0, acc11, false, false);
        }
        __syncthreads();                    // all reads of `cur` done before refill
    }

    // ---- store 16x16 C/D tiles: VGPR g -> row (khalf*8 + g), col l15 ----
    const int mB = blockM + waveM * 32;
    const int nB = blockN + waveN * 32;
    auto storeTile = [&](v8f acc, int mb, int nb) {
        const int row0 = mb + khalf * 8;
        const int col  = nb + l15;
        #pragma unroll
        for (int g = 0; g < 8; ++g) {
            float v = acc[g];
            if (relu) v = v > 0.0f ? v : 0.0f;
            C[(size_t)(row0 + g) * N + col] = v;
        }
    };
    storeTile(acc00, mB,      nB);
    storeTile(acc01, mB,      nB + 16);
    storeTile(acc10, mB + 16, nB);
    storeTile(acc11, mB + 16, nB + 16);
}

// ---------------------------------------------------------------------------
// GCN helper kernels
// ---------------------------------------------------------------------------
__global__ void deg_init(float* deg, int n) {
    int i = blockIdx.x * blockDim.x + threadIdx.x;
    if (i < n) deg[i] = 1.0f;                       // self-loop contribution
}

__global__ void deg_edges(const int* __restrict__ dst, float* deg, int e) {
    int i = blockIdx.x * blockDim.x + threadIdx.x;
    if (i < e) atomicAdd(&deg[dst[i]], 1.0f);
}

__global__ void deg_finish(const float* __restrict__ deg, float* dis, int n) {
    int i = blockIdx.x * blockDim.x + threadIdx.x;
    if (i < n) {
        float d = deg[i];
        dis[i] = d > 0.0f ? rsqrtf(d) : 0.0f;
    }
}

// Hl[i,:] = XW[i,:] * dis[i]^2   (self-loop message, atomics-free init)
__global__ void hl_init(const float* __restrict__ XW, const float* __restrict__ dis,
                        float* __restrict__ Hl, int total) {
    int i = blockIdx.x * blockDim.x + threadIdx.x;
    if (i < total) {
        int node = i >> 8;                          // FDIM == 256
        float s = dis[node];
        Hl[i] = XW[i] * s * s;
    }
}

// one thread per (edge, 4-feature chunk): Hl[dst] += XW[src] * dis[src]*dis[dst]
__global__ void edge_scatter(const int* __restrict__ src, const int* __restrict__ dst,
                             const float* __restrict__ dis, const float* __restrict__ XW,
                             float* __restrict__ Hl, int e) {
    long long gid = (long long)blockIdx.x * blockDim.x + threadIdx.x;
    int ei = (int)(gid >> 6);
    int c  = (int)(gid & 63) << 2;
    if (ei < e) {
        int s = src[ei], d = dst[ei];
        float ns = dis[s] * dis[d];
        const float* xr = XW + (size_t)s * FDIM + c;
        float* hr = Hl + (size_t)d * FDIM + c;
        v4f v = *(const v4f*)xr;
        atomicAdd(hr + 0, v.x * ns);
        atomicAdd(hr + 1, v.y * ns);
        atomicAdd(hr + 2, v.z * ns);
        atomicAdd(hr + 3, v.w * ns);
    }
}

// out = aL*(Hl + b_conv) + aH*Hh
__global__ void combine(const float* __restrict__ Hl, const float* __restrict__ Hh,
                        const float* __restrict__ bias, const float* __restrict__ aL,
                        const float* __restrict__ aH, float* __restrict__ out, int total) {
    int i = blockIdx.x * blockDim.x + threadIdx.x;
    if (i < total) {
        int f = i & (FDIM - 1);
        out[i] = aL[0] * (Hl[i] + bias[f]) + aH[0] * Hh[i];
    }
}

// ---------------------------------------------------------------------------
extern "C" void kernel_launch(void* const* d_in, const int* in_sizes, int n_in,
                              void* d_out, int out_size, void* d_ws, size_t ws_size,
                              hipStream_t stream) {
    const float* x      = (const float*)d_in[0];
    const int*   edges  = (const int*)d_in[1];      // [2, E] int32 (JAX x64 off)
    const float* lap    = (const float*)d_in[2];
    const float* d_inv  = (const float*)d_in[3];
    const float* W_high = (const float*)d_in[4];
    const float* W_conv = (const float*)d_in[5];
    const float* b_conv = (const float*)d_in[6];
    const float* aL     = (const float*)d_in[7];
    const float* aH     = (const float*)d_in[8];
    float* out = (float*)d_out;

    const int E  = in_sizes[1] / 2;
    const int NN = NNODES * FDIM;                   // 1,048,576 floats per matrix

    float* ws   = (float*)d_ws;
    float* bufA = ws;                               // R, then T2 = lap@T1
    float* bufB = bufA + NN;                        // T1 = d_inv@R, then Hh
    float* XW   = bufB + NN;                        // x @ W_conv
    float* Hl   = XW + NN;                          // GCN aggregation
    float* deg  = Hl + NN;                          // [4096]
    float* dis  = deg + NNODES;                     // [4096]

    const int* srcIdx = edges;
    const int* dstIdx = edges + E;

    dim3 blk(256);
    dim3 gemmGrid(NNODES / TILE_M, FDIM / TILE_N);  // 32 x 4

    // --- GCN degree / normalization ---
    deg_init<<<NNODES / 256, blk, 0, stream>>>(deg, NNODES);
    deg_edges<<<(E + 255) / 256, blk, 0, stream>>>(dstIdx, deg, E);
    deg_finish<<<NNODES / 256, blk, 0, stream>>>(deg, dis, NNODES);

    // --- small GEMMs (K=256) ---
    gemm_f32_wmma<<<gemmGrid, blk, 0, stream>>>(x, W_high, bufA, NNODES, FDIM, FDIM, 1); // R = relu(x@W_high)
    gemm_f32_wmma<<<gemmGrid, blk, 0, stream>>>(x, W_conv, XW,   NNODES, FDIM, FDIM, 0); // XW = x@W_conv

    // --- high-pass chain, reassociated: d_inv @ (lap @ (d_inv @ R)) ---
    gemm_f32_wmma<<<gemmGrid, blk, 0, stream>>>(d_inv, bufA, bufB, NNODES, FDIM, NNODES, 0); // T1
    gemm_f32_wmma<<<gemmGrid, blk, 0, stream>>>(lap,   bufB, bufA, NNODES, FDIM, NNODES, 0); // T2
    gemm_f32_wmma<<<gemmGrid, blk, 0, stream>>>(d_inv, bufA, bufB, NNODES, FDIM, NNODES, 0); // Hh

    // --- GCN aggregation ---
    hl_init<<<NN / 256, blk, 0, stream>>>(XW, dis, Hl, NN);
    {
        long long work = (long long)E * 64;         // 4-float chunks
        edge_scatter<<<(unsigned)((work + 255) / 256), blk, 0, stream>>>(srcIdx, dstIdx, dis, XW, Hl, E);
    }

    // --- final combine ---
    combine<<<NN / 256, blk, 0, stream>>>(Hl, bufB, b_conv, aL, aH, out, NN);
}